// Posterior_23768349016162
// MI455X (gfx1250) — compile-verified
//
#include <hip/hip_runtime.h>
#include <cmath>

#define R_ 64
#define K_ 32
#define D_ 512

typedef __attribute__((ext_vector_type(2))) float v2f;
typedef __attribute__((ext_vector_type(4))) float v4f;
typedef __attribute__((ext_vector_type(8))) float v8f;

#if defined(__HIP_DEVICE_COMPILE__) && __has_builtin(__builtin_amdgcn_global_load_async_to_lds_b32)
#define HAVE_ASYNC_LDS 1
#endif

// ---------------------------------------------------------------------------
// Kernel 1: one workgroup per (r,k) slice of W (1 MB each, 2048 slices).
// Computes t[rk] = u[rk] * tanh( x^T W_rk x + V_rk . x + b[rk] ).
// Roofline: 2.147 GB of W streamed once @ 23.3 TB/s => ~92us floor; so the
// main loop is coalesced b128 *non-temporal* loads of W (zero reuse, don't
// pollute L2) + fp32 VALU FMA, x staged in LDS via the gfx1250 async-to-LDS
// path, per-thread 4-wide x[e] chunk held in registers (LDS traffic = W/4).
// ---------------------------------------------------------------------------
__global__ __launch_bounds__(256) void qform_kernel(
    const float* __restrict__ x, const float* __restrict__ W,
    const float* __restrict__ V, const float* __restrict__ u,
    const float* __restrict__ b, float* __restrict__ t) {
  __shared__ __align__(16) float xs[D_];
  __shared__ float red[8];

  const int tid = threadIdx.x;
  const int rk  = blockIdx.x;

  // Stage x (2 KB) into LDS via gfx1250 GLOBAL_LOAD_ASYNC_TO_LDS_B32.
#ifdef HAVE_ASYNC_LDS
  for (int i = tid; i < D_; i += 256) {
    __builtin_amdgcn_global_load_async_to_lds_b32(
        (__attribute__((address_space(1))) int*)(uintptr_t)(x + i),
        (__attribute__((address_space(3))) int*)(xs + i),
        /*offset=*/0, /*cpol=*/0);
  }
#if __has_builtin(__builtin_amdgcn_s_wait_asynccnt)
  __builtin_amdgcn_s_wait_asynccnt(0);
#endif
#else
  for (int i = tid; i < D_; i += 256) xs[i] = x[i];
#endif
  __syncthreads();

  const float* __restrict__ Ws = W + (size_t)rk * (size_t)(D_ * D_);

  const int ecol = tid & 127;   // which 4-wide e-chunk this thread owns
  const int dgrp = tid >> 7;    // 0/1: even/odd d rows
  const v4f xe = *(const v4f*)(xs + ecol * 4);

  float acc = 0.0f;
  for (int d = dgrp; d < D_; d += 2) {
    // W is streamed exactly once: non-temporal b128 load.
    const v4f w = __builtin_nontemporal_load(
        (const v4f*)(Ws + (size_t)d * D_ + ecol * 4));
    if (d + 8 < D_) {
      // gfx1250 global_prefetch_b8: pull the row 4 iterations ahead.
      __builtin_prefetch(Ws + (size_t)(d + 8) * D_ + ecol * 4, 0, 0);
    }
    float s = fmaf(w[0], xe[0], fmaf(w[1], xe[1], fmaf(w[2], xe[2], w[3] * xe[3])));
    acc = fmaf(xs[d], s, acc);
  }

  // V . x contribution folded into the same partial sums.
  const float* __restrict__ Vs = V + (size_t)rk * D_;
  for (int d = tid; d < D_; d += 256) acc = fmaf(Vs[d], xs[d], acc);

  // wave32 reduction, then cross-wave via LDS.
  for (int off = 16; off > 0; off >>= 1) acc += __shfl_down(acc, off, 32);
  if ((tid & 31) == 0) red[tid >> 5] = acc;
  __syncthreads();

  if (tid == 0) {
    float pre = b[rk];
    for (int wv = 0; wv < 8; ++wv) pre += red[wv];
    t[rk] = u[rk] * tanhf(pre);
  }
}

// ---------------------------------------------------------------------------
// Kernel 2: scores[r] = sum_k t[r,k]; out[r] = sigmoid(scores[r]).
// Ones-matrix GEMM on the matrix unit:
//   D(16x16) = A(16x4 chunk of t) x B(4x16 ones) + C, accumulated over K.
// One full wave (EXEC all-1s, as WMMA requires). Emits v_wmma_f32_16x16x4_f32.
// A layout (32-bit 16x4): lane m(0..15): VGPR0=A[m,0], VGPR1=A[m,1];
//                         lane m+16:    VGPR0=A[m,2], VGPR1=A[m,3].
// C/D layout: lanes 0..15 hold M=v (VGPR v), lanes 16..31 hold M=v+8.
// ---------------------------------------------------------------------------
__global__ __launch_bounds__(32) void score_kernel(const float* __restrict__ t,
                                                   float* __restrict__ out) {
  const int lane = threadIdx.x;  // 0..31, exactly one wave
  const int m  = lane & 15;
  const int hi = lane >> 4;      // 0 -> K{0,1}, 1 -> K{2,3}

  v2f ones;
  ones[0] = 1.0f;
  ones[1] = 1.0f;

  for (int rc = 0; rc < 4; ++rc) {     // 4 chunks of 16 relations
    const int r = rc * 16 + m;
    v8f c = {};
    for (int kb = 0; kb < K_; kb += 4) {
      v2f a;
      a[0] = t[r * K_ + kb + hi * 2 + 0];
      a[1] = t[r * K_ + kb + hi * 2 + 1];
      c = __builtin_amdgcn_wmma_f32_16x16x4_f32(
          /*neg_a=*/false, a, /*neg_b=*/false, ones,
          /*c_mod=*/(short)0, c, /*reuse_a=*/false, /*reuse_b=*/false);
    }
    // Every output column holds the row-sums; read column N=0 from lane 0
    // (M=0..7) and lane 16 (M=8..15).
    if (m == 0) {
      const int rbase = rc * 16 + hi * 8;
      for (int v = 0; v < 8; ++v) {
        float s = c[v];
        out[rbase + v] = 1.0f / (1.0f + expf(-s));
      }
    }
  }
}

// ---------------------------------------------------------------------------
// Inputs (setup_inputs order): x[512], W[64,32,512,512], V[64,32,512],
// u[64,32], b[64,32]. Output: 64 floats. Workspace: t[2048] floats.
// ---------------------------------------------------------------------------
extern "C" void kernel_launch(void* const* d_in, const int* in_sizes, int n_in,
                              void* d_out, int out_size, void* d_ws, size_t ws_size,
                              hipStream_t stream) {
  const float* x = (const float*)d_in[0];
  const float* W = (const float*)d_in[1];
  const float* V = (const float*)d_in[2];
  const float* u = (const float*)d_in[3];
  const float* b = (const float*)d_in[4];
  float* t = (float*)d_ws;  // R_*K_ = 2048 floats of scratch

  qform_kernel<<<R_ * K_, 256, 0, stream>>>(x, W, V, u, b, t);
  score_kernel<<<1, 32, 0, stream>>>(t, (float*)d_out);
}